// STDPCoordinationSystem_73435350827443
// MI455X (gfx1250) — compile-verified
//
#include <hip/hip_runtime.h>
#include <hip/hip_bf16.h>

// ---------------------------------------------------------------------------
// Problem constants (from the reference)
// ---------------------------------------------------------------------------
#define NB 64
#define NA 32
#define BOARD 64
#define ND 5
#define NAGENT (NB * NA)                 // 2048
#define CELLS (BOARD * BOARD)            // 4096
#define FAC_ELEMS ((size_t)NAGENT * CELLS * ND)   // 41,943,040
#define CH_ELEMS  ((size_t)NAGENT * CELLS)        // 8,388,608

#define DECAY_FAC 0.98019867f   // exp(-1/50)
#define DECAY_INH 0.95122945f   // exp(-1/20)
#define DECAY_CH  0.9f          // 1 - 1/10
#define LR_FAC 0.0002f
#define LR_INH 0.0001f
#define SAFE_THRESH 0.7f
#define ALIGN_THRESH 0.6f

typedef __attribute__((ext_vector_type(2))) float v2f;
typedef __attribute__((ext_vector_type(4))) float vf4;
typedef __attribute__((ext_vector_type(8))) float v8f;

// ---------------------------------------------------------------------------
// Phase A: per-agent prep + WMMA for the [2048x2] @ [2x5] alignment matmul.
// One wave (32 lanes) per block handles 16 agents via one
// V_WMMA_F32_16X16X4_F32 (M=16 agents, K=2 live of 4, N=5 live of 16).
// ---------------------------------------------------------------------------
__global__ __launch_bounds__(32)
void stdp_prep_wmma(const float* __restrict__ positions,
                    const float* __restrict__ goals,
                    const float* __restrict__ spikes,
                    const float* __restrict__ ch_in,
                    const int*   __restrict__ collisions,
                    float* __restrict__ safety_out,
                    int*   __restrict__ ws_idx,
                    float* __restrict__ ws_chadd,
                    float* __restrict__ ws_fac,
                    float* __restrict__ ws_inh)
{
    const int lane = threadIdx.x;          // 0..31
    const int base = blockIdx.x * 16;      // first agent of this wave's tile

    float gx = 0.f, gy = 0.f, safety = 0.f;
    if (lane < 16) {
        const int agent = base + lane;
        const float px = positions[agent * 2 + 0];
        const float py = positions[agent * 2 + 1];
        int x = (int)px; x = x < 0 ? 0 : (x > BOARD - 1 ? BOARD - 1 : x);
        int y = (int)py; y = y < 0 ? 0 : (y > BOARD - 1 ? BOARD - 1 : y);
        const int cell = y * BOARD + x;

        const float gvx = goals[agent * 2 + 0] - px;
        const float gvy = goals[agent * 2 + 1] - py;
        const float inv = 1.f / (sqrtf(gvx * gvx + gvy * gvy) + 1e-8f);
        gx = gvx * inv;
        gy = gvy * inv;

        const float coll = (collisions[agent] != 0) ? 1.f : 0.f;
        const float chv  = ch_in[(size_t)agent * CELLS + cell] * DECAY_CH + coll * 0.1f;
        float s = 1.f - fminf(fmaxf(chv, 0.f), 1.f);
        safety = s;

        safety_out[agent] = s;
        ws_idx[agent]     = cell;
        ws_chadd[agent]   = coll * 0.1f;
    }

    // A matrix 16x4 (MxK), f32: VGPR0 = {lanes0-15: K=0, lanes16-31: K=2},
    // VGPR1 = {K=1, K=3}.  K=2,3 are zero padding.
    v2f a;
    a.x = (lane < 16) ? gx : 0.f;
    a.y = (lane < 16) ? gy : 0.f;

    // B matrix 4x16 (KxN): rows striped across lanes within a VGPR
    // (row k in lanes 0-15 of VGPR k, row k+2 in lanes 16-31).
    // Rows: K=0 -> delta.x per direction, K=1 -> delta.y; K=2,3 zero.
    // deltas = {(0,0),(1,0),(0,1),(-1,0),(0,-1)} for N=0..4, zero beyond.
    float bx = 0.f, by = 0.f;
    if (lane < 16) {
        bx = (lane == 1) ? 1.f : ((lane == 3) ? -1.f : 0.f);
        by = (lane == 2) ? 1.f : ((lane == 4) ? -1.f : 0.f);
    }
    v2f b; b.x = bx; b.y = by;

    v8f c = {};
    // D[m][n] = gvn[m] . delta[n]
    v8f d = __builtin_amdgcn_wmma_f32_16x16x4_f32(
        /*neg_a=*/false, a, /*neg_b=*/false, b,
        /*c_mod=*/(short)0, c, /*reuse_a=*/false, /*reuse_b=*/false);

    // D layout: lane holds column N = lane%16 for rows M = r + 8*(lane>=16).
    const int  n      = lane & 15;
    const int  mbase  = (lane >= 16) ? 8 : 0;
    const bool active = (n < ND);

#pragma unroll
    for (int r = 0; r < 8; ++r) {
        const int m = mbase + r;
        // all lanes participate in the shuffle (safety lives in lanes 0..15)
        const float sa = __shfl(safety, m, 32);
        if (active) {
            const int   agent = base + m;
            const float align = 0.5f * d[r] + 0.5f;       // (dot + 1) / 2
            const float sp    = spikes[agent * ND + n];
            const bool  spike = sp > 0.5f;
            const bool  safe  = sa > SAFE_THRESH;
            const float fa = (spike && safe && (align > ALIGN_THRESH)) ? LR_FAC : 0.f;
            const float ia = (spike && (!safe || (align <= 1.0f - ALIGN_THRESH))) ? LR_INH : 0.f;
            ws_fac[agent * ND + n] = fa;
            ws_inh[agent * ND + n] = ia;
        }
    }
}

// ---------------------------------------------------------------------------
// Phase B: bandwidth-bound stream  out = in * scale  (128-bit, non-temporal).
// Data is streamed exactly once and exceeds L2 (740MB > 192MB), so NT hints.
// ---------------------------------------------------------------------------
__global__ __launch_bounds__(256)
void stdp_scale_stream(const vf4* __restrict__ in, vf4* __restrict__ out,
                       float scale, int n4)
{
    int i = blockIdx.x * blockDim.x + threadIdx.x;
    const int stride = gridDim.x * blockDim.x;
    for (; i < n4; i += stride) {
        vf4 v = __builtin_nontemporal_load(&in[i]);
        v *= scale;
        __builtin_nontemporal_store(v, &out[i]);
    }
}

// ---------------------------------------------------------------------------
// Phase C: scatter fix-up after the stream.  Each agent owns a disjoint
// [CELLS x ND] slice, so plain read-modify-write is race-free.
// ---------------------------------------------------------------------------
__global__ __launch_bounds__(256)
void stdp_scatter(const int*   __restrict__ ws_idx,
                  const float* __restrict__ ws_chadd,
                  const float* __restrict__ ws_fac,
                  const float* __restrict__ ws_inh,
                  float* __restrict__ out_fac,
                  float* __restrict__ out_inh,
                  float* __restrict__ out_ch)
{
    const int agent = blockIdx.x * blockDim.x + threadIdx.x;
    if (agent >= NAGENT) return;
    const int    cell  = ws_idx[agent];
    const size_t fbase = ((size_t)agent * CELLS + cell) * ND;
#pragma unroll
    for (int dd = 0; dd < ND; ++dd) {
        out_fac[fbase + dd] += ws_fac[agent * ND + dd];
        out_inh[fbase + dd] += ws_inh[agent * ND + dd];
    }
    out_ch[(size_t)agent * CELLS + cell] += ws_chadd[agent];
}

// ---------------------------------------------------------------------------
// Launch
// ---------------------------------------------------------------------------
extern "C" void kernel_launch(void* const* d_in, const int* in_sizes, int n_in,
                              void* d_out, int out_size, void* d_ws, size_t ws_size,
                              hipStream_t stream)
{
    (void)in_sizes; (void)n_in; (void)out_size; (void)ws_size;

    const float* positions  = (const float*)d_in[0];
    const float* goals      = (const float*)d_in[1];
    const float* spikes     = (const float*)d_in[2];
    const float* fac_in     = (const float*)d_in[3];
    const float* inh_in     = (const float*)d_in[4];
    const float* ch_in      = (const float*)d_in[5];
    const int*   collisions = (const int*)d_in[6];

    float* out_fac    = (float*)d_out;                       // 41,943,040
    float* out_inh    = out_fac + FAC_ELEMS;                 // 41,943,040
    float* out_ch     = out_inh + FAC_ELEMS;                 // 8,388,608
    float* out_safety = out_ch  + CH_ELEMS;                  // 2,048

    // Workspace carve-out (98,304 bytes total)
    char* ws = (char*)d_ws;
    int*   ws_idx   = (int*)ws;                              // 2048 ints
    float* ws_chadd = (float*)(ws + NAGENT * sizeof(int));   // 2048 f32
    float* ws_fac   = ws_chadd + NAGENT;                     // 2048*5 f32
    float* ws_inh   = ws_fac + NAGENT * ND;                  // 2048*5 f32

    // Phase A: 128 waves of 32 lanes, 16 agents each (one WMMA per wave).
    stdp_prep_wmma<<<NAGENT / 16, 32, 0, stream>>>(
        positions, goals, spikes, ch_in, collisions,
        out_safety, ws_idx, ws_chadd, ws_fac, ws_inh);

    // Phase B: three NT streaming passes (the ~740MB roofline part).
    {
        const int n4_fac = (int)(FAC_ELEMS / 4);   // 10,485,760
        const int n4_ch  = (int)(CH_ELEMS / 4);    //  2,097,152
        int blk_fac = (n4_fac + 255) / 256; if (blk_fac > 40960) blk_fac = 40960;
        int blk_ch  = (n4_ch  + 255) / 256; if (blk_ch  > 40960) blk_ch  = 40960;

        stdp_scale_stream<<<blk_fac, 256, 0, stream>>>(
            (const vf4*)fac_in, (vf4*)out_fac, DECAY_FAC, n4_fac);
        stdp_scale_stream<<<blk_fac, 256, 0, stream>>>(
            (const vf4*)inh_in, (vf4*)out_inh, DECAY_INH, n4_fac);
        stdp_scale_stream<<<blk_ch, 256, 0, stream>>>(
            (const vf4*)ch_in, (vf4*)out_ch, DECAY_CH, n4_ch);
    }

    // Phase C: scatter-add fix-up (ordered after B on the same stream).
    stdp_scatter<<<(NAGENT + 255) / 256, 256, 0, stream>>>(
        ws_idx, ws_chadd, ws_fac, ws_inh, out_fac, out_inh, out_ch);
}